// PPYoloELoss_31576599560730
// MI455X (gfx1250) — compile-verified
//
#include <hip/hip_runtime.h>

#define TOPK 13

typedef __attribute__((ext_vector_type(2))) float v2f;
typedef __attribute__((ext_vector_type(8))) float v8f;

// ---------------------------------------------------------------------------
// Wave32 sum reduction on the CDNA5 matrix pipe.
// Stage 1: A(16x4) holds lane partials (vgpr0=K0 lanes0-15, K2 lanes16-31),
//          B = ones(4x16)  ->  D[m,n] = v[m] + v[m+16]  (row-constant).
// Stage 2: fold the 16 row sums with 4 accumulating WMMAs (pairs of D vgprs
//          re-enter as A).  Result: full wave sum replicated in all lanes.
// EXEC must be all ones at the call site (guaranteed: full waves, reconverged).
// ---------------------------------------------------------------------------
__device__ __forceinline__ float wave_reduce_sum(float v) {
  v2f a;    a[0] = v;    a[1] = 0.0f;
  v2f ones; ones[0] = 1.0f; ones[1] = 1.0f;
  v8f c = {};
  c = __builtin_amdgcn_wmma_f32_16x16x4_f32(false, a, false, ones, (short)0, c, false, false);
  v8f acc = {};
  v2f a2;
  a2[0] = c[0]; a2[1] = c[1];
  acc = __builtin_amdgcn_wmma_f32_16x16x4_f32(false, a2, false, ones, (short)0, acc, false, false);
  a2[0] = c[2]; a2[1] = c[3];
  acc = __builtin_amdgcn_wmma_f32_16x16x4_f32(false, a2, false, ones, (short)0, acc, false, false);
  a2[0] = c[4]; a2[1] = c[5];
  acc = __builtin_amdgcn_wmma_f32_16x16x4_f32(false, a2, false, ones, (short)0, acc, false, false);
  a2[0] = c[6]; a2[1] = c[7];
  acc = __builtin_amdgcn_wmma_f32_16x16x4_f32(false, a2, false, ones, (short)0, acc, false, false);
  return acc[0];
}

__device__ __forceinline__ float iou_xyxy(float4 a, float4 b) {
  float x1 = fmaxf(a.x, b.x), y1 = fmaxf(a.y, b.y);
  float x2 = fminf(a.z, b.z), y2 = fminf(a.w, b.w);
  float ov = fmaxf(x2 - x1, 0.f) * fmaxf(y2 - y1, 0.f);
  float a1 = fmaxf(a.z - a.x, 0.f) * fmaxf(a.w - a.y, 0.f);
  float a2 = fmaxf(b.z - b.x, 0.f) * fmaxf(b.w - b.y, 0.f);
  return ov / (a1 + a2 - ov + 1e-9f);
}

__device__ __forceinline__ float giou_loss_f(float4 p, float4 g) {
  float x1 = fmaxf(p.x, g.x), y1 = fmaxf(p.y, g.y);
  float x2 = fminf(p.z, g.z), y2 = fminf(p.w, g.w);
  float ov = fmaxf(x2 - x1, 0.f) * fmaxf(y2 - y1, 0.f);
  float ap = fmaxf(p.z - p.x, 0.f) * fmaxf(p.w - p.y, 0.f);
  float ag = fmaxf(g.z - g.x, 0.f) * fmaxf(g.w - g.y, 0.f);
  float un = ap + ag - ov + 1e-10f;
  float iou = ov / un;
  float cx1 = fminf(p.x, g.x), cy1 = fminf(p.y, g.y);
  float cx2 = fmaxf(p.z, g.z), cy2 = fmaxf(p.w, g.w);
  float c = fmaxf(cx2 - cx1, 0.f) * fmaxf(cy2 - cy1, 0.f) + 1e-10f;
  return 1.0f - (iou - (c - un) / c);
}

// ---------------------------------------------------------------------------
__global__ void zero_kernel(unsigned* __restrict__ p, size_t n) {
  size_t i = (size_t)blockIdx.x * blockDim.x + threadIdx.x;
  if (i < n) p[i] = 0u;
}

// ---------------------------------------------------------------------------
// One block per (b,n): exact top-13 of metric over L anchors (tie-break:
// lowest index, matching jax.lax.top_k).  Result = bitmask over anchors.
// ---------------------------------------------------------------------------
__global__ void __launch_bounds__(256)
topk_kernel(const float* __restrict__ ps, const float* __restrict__ pb,
            const float* __restrict__ ap, const int* __restrict__ gl,
            const float* __restrict__ gb, const float* __restrict__ pad,
            unsigned* __restrict__ bits, int L, int C, int N, int W32) {
  int bn = blockIdx.x;
  if (pad[bn] == 0.0f) return;  // padded GT: is_in_topk is zeroed by pad later
  int b = bn / N;
  __shared__ float4 sgt;
  __shared__ int sgl;
  if (threadIdx.x == 0) { sgt = ((const float4*)gb)[bn]; sgl = gl[bn]; }
  __syncthreads();
  float4 g = sgt;
  int lab = sgl;

  float tv[TOPK]; int ti[TOPK];
#pragma unroll
  for (int k = 0; k < TOPK; ++k) { tv[k] = -1.0f; ti[k] = 0x7fffffff; }

  const float4* pbb = (const float4*)pb + (size_t)b * L;
  const float2* apv = (const float2*)ap;
  const float* psb = ps + (size_t)b * L * C;

  for (int l = threadIdx.x; l < L; l += blockDim.x) {
    float4 p = pbb[l];
    float iou = iou_xyxy(g, p);
    float2 a = apv[l];
    float din = fminf(fminf(a.x - g.x, a.y - g.y), fminf(g.z - a.x, g.w - a.y));
    float ing = (din > 1e-9f) ? 1.0f : 0.0f;
    float sc = psb[(size_t)l * C + lab];
    float i2 = iou * iou;
    float m = sc * (i2 * i2 * i2) * ing;
    if (m > tv[TOPK - 1]) {  // strict >: equal values keep earlier index ahead
#pragma unroll
      for (int j = TOPK - 1; j >= 0; --j) {
        if (m > tv[j]) {
          if (j < TOPK - 1) { tv[j + 1] = tv[j]; ti[j + 1] = ti[j]; }
          tv[j] = m; ti[j] = l;
        } else break;
      }
    }
  }

  __shared__ float sv[256 * TOPK];
  __shared__ int si[256 * TOPK];
  int t = threadIdx.x;
#pragma unroll
  for (int k = 0; k < TOPK; ++k) { sv[t * TOPK + k] = tv[k]; si[t * TOPK + k] = ti[k]; }
  __syncthreads();

  for (int s = blockDim.x >> 1; s >= 1; s >>= 1) {
    if (t < s) {
      float ov[TOPK]; int oi[TOPK];
      int i = 0, j = 0;
#pragma unroll
      for (int k = 0; k < TOPK; ++k) {
        float va = sv[t * TOPK + i];       int ia = si[t * TOPK + i];
        float vb = sv[(t + s) * TOPK + j]; int ib = si[(t + s) * TOPK + j];
        bool takeA = (va > vb) || (va == vb && ia <= ib);
        if (takeA) { ov[k] = va; oi[k] = ia; ++i; }
        else       { ov[k] = vb; oi[k] = ib; ++j; }
      }
#pragma unroll
      for (int k = 0; k < TOPK; ++k) { sv[t * TOPK + k] = ov[k]; si[t * TOPK + k] = oi[k]; }
    }
    __syncthreads();
  }

  if (t == 0) {
    unsigned* row = bits + (size_t)bn * W32;
#pragma unroll
    for (int k = 0; k < TOPK; ++k) {
      int idx = si[k];
      row[idx >> 5] |= (1u << (idx & 31));
    }
  }
}

// ---------------------------------------------------------------------------
// One thread per (b,l): resolve assignment, record align/iou, push per-GT
// maxima with order-independent atomicMax (float-as-uint; values >= 0).
// ---------------------------------------------------------------------------
__global__ void __launch_bounds__(256)
assign_kernel(const float* __restrict__ ps, const float* __restrict__ pb,
              const float* __restrict__ ap, const int* __restrict__ gl,
              const float* __restrict__ gb, const float* __restrict__ pad,
              const unsigned* __restrict__ bits,
              float* __restrict__ max_met, float* __restrict__ max_iou,
              int* __restrict__ assigned_n, float* __restrict__ align_val,
              float* __restrict__ iou_val, int L, int C, int N, int W32) {
  int b = blockIdx.y;
  int l = blockIdx.x * blockDim.x + threadIdx.x;
  __shared__ float4 sgb[128];
  __shared__ int sgl[128];
  __shared__ float spad[128];
  for (int i = threadIdx.x; i < N; i += blockDim.x) {
    sgb[i] = ((const float4*)gb)[(size_t)b * N + i];
    sgl[i] = gl[b * N + i];
    spad[i] = pad[b * N + i];
  }
  __syncthreads();
  if (l >= L) return;

  float4 p = ((const float4*)pb)[(size_t)b * L + l];
  float2 a = ((const float2*)ap)[l];
  int word = l >> 5;
  unsigned bit = 1u << (l & 31);

  float best_iou = -1.0f;
  int best_n = 0, cnt = 0, first_n = -1;
  for (int n = 0; n < N; ++n) {
    float4 g = sgb[n];
    float iou = iou_xyxy(g, p);
    if (iou > best_iou) { best_iou = iou; best_n = n; }  // first-max, incl. pads
    unsigned w = bits[(size_t)(b * N + n) * W32 + word];
    float din = fminf(fminf(a.x - g.x, a.y - g.y), fminf(g.z - a.x, g.w - a.y));
    bool m = ((w & bit) != 0) && (din > 1e-9f) && (spad[n] > 0.0f);
    if (m) { ++cnt; if (first_n < 0) first_n = n; }
  }
  int ns = (cnt > 1) ? best_n : ((cnt == 1) ? first_n : -1);

  size_t idx = (size_t)b * L + l;
  float av = 0.0f, iv = 0.0f;
  if (ns >= 0) {
    float4 g = sgb[ns];
    float iou = iou_xyxy(g, p);
    float sc = ps[idx * C + sgl[ns]];
    float i2 = iou * iou;
    av = sc * (i2 * i2 * i2);  // align (no in_gts factor, per reference)
    iv = iou;
    atomicMax((unsigned*)&max_met[b * N + ns], __float_as_uint(av));
    atomicMax((unsigned*)&max_iou[b * N + ns], __float_as_uint(iv));
  }
  assigned_n[idx] = ns;
  align_val[idx] = av;
  iou_val[idx] = iv;
}

// ---------------------------------------------------------------------------
// One thread per (b,l): VFL BCE over 80 classes + GIoU + asc sum; wave
// reduction rides the WMMA pipe; fixed-order block combine (deterministic).
// ---------------------------------------------------------------------------
__global__ void __launch_bounds__(256)
loss_kernel(const float* __restrict__ ps, const float* __restrict__ pb,
            const int* __restrict__ gl, const float* __restrict__ gb,
            const int* __restrict__ assigned_n, const float* __restrict__ align_val,
            const float* __restrict__ max_met, const float* __restrict__ max_iou,
            float* __restrict__ part_cls, float* __restrict__ part_iou,
            float* __restrict__ part_asc, int L, int C, int N, int total) {
  int tid = blockIdx.x * blockDim.x + threadIdx.x;
  float pc = 0.f, pi = 0.f, pa = 0.f;
  if (tid < total) {
    int b = tid / L;
    size_t idx = (size_t)tid;
    int ns = assigned_n[idx];
    float scale = 0.f;
    int a = -1;
    if (ns >= 0) {
      scale = align_val[idx] / (max_met[b * N + ns] + 1e-9f) * max_iou[b * N + ns];
      a = gl[b * N + ns];
    }
    const float* row = ps + idx * C;
    for (int c = 0; c < C; ++c) {
      float p = row[c];
      p = fminf(fmaxf(p, 1e-9f), 1.0f - 1e-9f);
      float l1m = log1pf(-p);
      if (c == a) {
        float bce = -(scale * logf(p) + (1.0f - scale) * l1m);
        pc += bce * scale;                 // w = asc at the target channel
      } else {
        pc += 0.75f * p * p * (-l1m);      // w = alpha * p^gamma, bce = -log(1-p)
      }
    }
    pa = scale;
    if (ns >= 0) {
      float4 pbx = ((const float4*)pb)[idx];
      float4 g = ((const float4*)gb)[(size_t)b * N + ns];
      pi = giou_loss_f(pbx, g) * scale;    // bbox_weight = asc.sum(-1) * pos
    }
  }
  // Reconverged here: EXEC all ones -> WMMA-legal.
  float s1 = wave_reduce_sum(pc);
  float s2 = wave_reduce_sum(pi);
  float s3 = wave_reduce_sum(pa);
  __shared__ float b1[8], b2[8], b3[8];
  int wv = threadIdx.x >> 5;
  if ((threadIdx.x & 31) == 0) { b1[wv] = s1; b2[wv] = s2; b3[wv] = s3; }
  __syncthreads();
  if (threadIdx.x == 0) {
    float a1 = 0.f, a2 = 0.f, a3 = 0.f;
    int nw = blockDim.x >> 5;
    for (int w = 0; w < nw; ++w) { a1 += b1[w]; a2 += b2[w]; a3 += b3[w]; }
    part_cls[blockIdx.x] = a1;
    part_iou[blockIdx.x] = a2;
    part_asc[blockIdx.x] = a3;
  }
}

__global__ void __launch_bounds__(32)
finalize_kernel(const float* __restrict__ part_cls, const float* __restrict__ part_iou,
                const float* __restrict__ part_asc, int nblk, float* __restrict__ out) {
  int t = threadIdx.x;
  float a = 0.f, b = 0.f, c = 0.f;
  for (int i = t; i < nblk; i += 32) { a += part_cls[i]; b += part_iou[i]; c += part_asc[i]; }
  a = wave_reduce_sum(a);
  b = wave_reduce_sum(b);
  c = wave_reduce_sum(c);
  if (t == 0) out[0] = (1.0f * a + 2.5f * b) / fmaxf(c, 1.0f);
}

// ---------------------------------------------------------------------------
extern "C" void kernel_launch(void* const* d_in, const int* in_sizes, int n_in,
                              void* d_out, int out_size, void* d_ws, size_t ws_size,
                              hipStream_t stream) {
  const float* ps  = (const float*)d_in[0];   // pred_scores   [B,L,C]
  const float* pb  = (const float*)d_in[1];   // pred_bboxes   [B,L,4]
  const float* ap  = (const float*)d_in[2];   // anchor_points [L,2]
  const int*   gl  = (const int*)d_in[3];     // gt_labels     [B,N,1]
  const float* gb  = (const float*)d_in[4];   // gt_bboxes     [B,N,4]
  const float* pad = (const float*)d_in[5];   // pad_gt_mask   [B,N,1]

  int L = in_sizes[2] / 2;
  int B = in_sizes[1] / (4 * L);
  int C = in_sizes[0] / (B * L);
  int N = in_sizes[3] / B;
  int W32 = (L + 31) / 32;
  int total = B * L;
  int nblk3 = (total + 255) / 256;

  char* w = (char*)d_ws;
  unsigned* bits    = (unsigned*)w; w += (size_t)B * N * W32 * 4;
  float* max_met    = (float*)w;    w += (size_t)B * N * 4;
  float* max_iou    = (float*)w;    w += (size_t)B * N * 4;
  int*   assigned_n = (int*)w;      w += (size_t)total * 4;
  float* align_val  = (float*)w;    w += (size_t)total * 4;
  float* iou_val    = (float*)w;    w += (size_t)total * 4;
  float* part_cls   = (float*)w;    w += (size_t)nblk3 * 4;
  float* part_iou   = (float*)w;    w += (size_t)nblk3 * 4;
  float* part_asc   = (float*)w;    w += (size_t)nblk3 * 4;

  // bits + max_met + max_iou are contiguous: one zero pass (re-zero every call).
  size_t nz = (size_t)B * N * (W32 + 2);
  zero_kernel<<<(unsigned)((nz + 255) / 256), 256, 0, stream>>>(bits, nz);

  topk_kernel<<<B * N, 256, 0, stream>>>(ps, pb, ap, gl, gb, pad, bits, L, C, N, W32);

  dim3 g2((L + 255) / 256, B);
  assign_kernel<<<g2, 256, 0, stream>>>(ps, pb, ap, gl, gb, pad, bits, max_met, max_iou,
                                        assigned_n, align_val, iou_val, L, C, N, W32);

  loss_kernel<<<nblk3, 256, 0, stream>>>(ps, pb, gl, gb, assigned_n, align_val,
                                         max_met, max_iou, part_cls, part_iou, part_asc,
                                         L, C, N, total);

  finalize_kernel<<<1, 32, 0, stream>>>(part_cls, part_iou, part_asc, nblk3, (float*)d_out);
}